// GraphLearningModule_BernVAE_32512902431526
// MI455X (gfx1250) — compile-verified
//
#include <hip/hip_runtime.h>
#include <hip/hip_bf16.h>
#include <math.h>

typedef __attribute__((ext_vector_type(2))) float v2f;
typedef __attribute__((ext_vector_type(8))) float v8f;

#define BB    4
#define NN    512
#define TT    288
#define LL    128
#define MM    (BB * NN)   // 2048 flattened rows

// -------------------------------------------------------------------------
// Kernel 1: x1 = leaky(x @ W1^T + b1), x2 = leaky(x @ W2^T + b2)
// GEMM M=2048, K=288, Nout=128.  One wave32 per 16x16 output tile, f32 WMMA,
// K stepped by 4 (72 v_wmma_f32_16x16x4_f32 per wave).
// blockIdx.z selects (W1->x1) vs (W2->x2).
// -------------------------------------------------------------------------
__global__ __launch_bounds__(128) void bernvae_in_gemm(
    const float* __restrict__ x,
    const float* __restrict__ W1, const float* __restrict__ W2,
    const float* __restrict__ b1, const float* __restrict__ b2,
    float* __restrict__ o1, float* __restrict__ o2)
{
    const int ln    = threadIdx.x;                 // lane 0..31
    const int mt    = blockIdx.x * 4 + threadIdx.y; // M tile 0..127
    const int lt    = blockIdx.y;                  // L tile 0..7
    const int which = blockIdx.z;

    const float* W    = which ? W2 : W1;
    const float* bias = which ? b2 : b1;
    float*       out  = which ? o2 : o1;

    const int half = ln >> 4;   // 0: K=k,k+1  1: K=k+2,k+3
    const int lid  = ln & 15;
    const int m0 = mt * 16, l0 = lt * 16;

    const float* arow = x + (size_t)(m0 + lid) * TT;  // A row m0+lid
    const float* brow = W + (size_t)(l0 + lid) * TT;  // B^T: W[l, t]

    v8f c = {};
    for (int k = 0; k < TT; k += 4) {
        const int ka = k + half * 2;               // 8-byte aligned (ka even)
        v2f a = *(const v2f*)(arow + ka);
        v2f b = *(const v2f*)(brow + ka);
        c = __builtin_amdgcn_wmma_f32_16x16x4_f32(
                false, a, false, b, (short)0, c, false, false);
    }

    const float bv = bias[l0 + lid];
    #pragma unroll
    for (int r = 0; r < 8; ++r) {
        const int row = half * 8 + r;              // C layout: M = r + 8*(lane/16)
        float v = c[r] + bv;
        v = (v >= 0.0f) ? v : 0.2f * v;            // leaky
        out[(size_t)(m0 + row) * LL + l0 + lid] = v;
    }
}

// -------------------------------------------------------------------------
// Kernel 2: s1 = x1 @ Wp[:, :L]^T ;  s2p = x2 @ Wp[:, L:]^T + bp
// GEMM M=2048, K=128, Nout=128.  bp folded into s2 so the pairwise kernel
// only needs s1 + s2p.
// -------------------------------------------------------------------------
__global__ __launch_bounds__(128) void bernvae_lat_gemm(
    const float* __restrict__ x1, const float* __restrict__ x2,
    const float* __restrict__ Wp, const float* __restrict__ bp,
    float* __restrict__ s1, float* __restrict__ s2p)
{
    const int ln    = threadIdx.x;
    const int mt    = blockIdx.x * 4 + threadIdx.y;
    const int lt    = blockIdx.y;
    const int which = blockIdx.z;

    const float* A     = which ? x2 : x1;
    const float* Wbase = Wp + which * LL;          // Wp is (L, 2L) row-major
    float*       out   = which ? s2p : s1;

    const int half = ln >> 4;
    const int lid  = ln & 15;
    const int m0 = mt * 16, l0 = lt * 16;

    const float* arow = A     + (size_t)(m0 + lid) * LL;
    const float* brow = Wbase + (size_t)(l0 + lid) * (2 * LL);

    v8f c = {};
    for (int k = 0; k < LL; k += 4) {
        const int ka = k + half * 2;
        v2f a = *(const v2f*)(arow + ka);
        v2f b = *(const v2f*)(brow + ka);
        c = __builtin_amdgcn_wmma_f32_16x16x4_f32(
                false, a, false, b, (short)0, c, false, false);
    }

    const float bv = which ? bp[l0 + lid] : 0.0f;
    #pragma unroll
    for (int r = 0; r < 8; ++r) {
        const int row = half * 8 + r;
        out[(size_t)(m0 + row) * LL + l0 + lid] = c[r] + bv;
    }
}

// -------------------------------------------------------------------------
// Kernel 3: fused pairwise + head + Concrete relaxation.
//   logits[b,i,j] = sum_l Wb[l]*leaky(s1[b,i,l]+s2p[b,j,l]) + bb
//   p[i,j] = mean_b sigmoid(logits); zero diagonal
//   adj    = sigmoid((clip(logit(p)) + logit(noise)) / 0.2)
// 16x16 (i,j) tile per 256-thread block; s1/s2 rows staged in padded LDS
// (stride 132 floats -> no 16-way bank conflicts on the j-stride reads).
// -------------------------------------------------------------------------
__global__ __launch_bounds__(256) void bernvae_pairwise(
    const float* __restrict__ s1, const float* __restrict__ s2p,
    const float* __restrict__ Wb, const float* __restrict__ bb,
    const float* __restrict__ noise, float* __restrict__ adj)
{
    __shared__ float S1[16][132];
    __shared__ float S2[16][132];
    __shared__ float Wbs[LL];

    const int tx  = threadIdx.x;         // j within tile
    const int ty  = threadIdx.y;         // i within tile
    const int tid = ty * 16 + tx;
    const int i0  = blockIdx.y * 16;
    const int j0  = blockIdx.x * 16;

    if (tid < LL) Wbs[tid] = Wb[tid];
    const float bbv = bb[0];

    float psum = 0.0f;
    for (int b = 0; b < BB; ++b) {
        __syncthreads();                 // also covers Wbs on first iter
        for (int idx = tid; idx < 16 * LL; idx += 256) {
            const int r = idx >> 7, l = idx & (LL - 1);
            S1[r][l] = s1 [((size_t)b * NN + i0 + r) * LL + l];
            S2[r][l] = s2p[((size_t)b * NN + j0 + r) * LL + l];
        }
        __syncthreads();

        float acc = bbv;
        #pragma unroll 4
        for (int l = 0; l < LL; ++l) {
            float v = S1[ty][l] + S2[tx][l];
            v = (v >= 0.0f) ? v : 0.2f * v;
            acc = fmaf(Wbs[l], v, acc);
        }
        psum += 1.0f / (1.0f + expf(-acc));
    }

    float p = psum * 0.25f;
    const int i = i0 + ty, j = j0 + tx;
    if (i == j) p = 0.0f;                // diagonal -> logit clamps to -10

    float lp = logf(p + 1e-10f) - log1pf(-p + 1e-10f);
    lp = fminf(10.0f, fmaxf(-10.0f, lp));

    const float ns = noise[(size_t)i * NN + j];
    const float lg = logf(ns) - log1pf(-ns);

    adj[(size_t)i * NN + j] = 1.0f / (1.0f + expf(-(lp + lg) * 5.0f));
}

// -------------------------------------------------------------------------
extern "C" void kernel_launch(void* const* d_in, const int* in_sizes, int n_in,
                              void* d_out, int out_size, void* d_ws, size_t ws_size,
                              hipStream_t stream) {
    const float* x     = (const float*)d_in[0];
    const float* W1    = (const float*)d_in[1];
    const float* b1    = (const float*)d_in[2];
    const float* W2    = (const float*)d_in[3];
    const float* b2    = (const float*)d_in[4];
    const float* Wp    = (const float*)d_in[5];
    const float* bp    = (const float*)d_in[6];
    const float* Wb    = (const float*)d_in[7];
    const float* bb    = (const float*)d_in[8];
    const float* noise = (const float*)d_in[9];
    float* out = (float*)d_out;

    float* ws  = (float*)d_ws;
    float* x1  = ws;                    // 2048*128 each = 1 MB
    float* x2  = ws + (size_t)1 * MM * LL;
    float* s1  = ws + (size_t)2 * MM * LL;
    float* s2p = ws + (size_t)3 * MM * LL;

    dim3 gGemm(MM / 64, LL / 16, 2);    // (32, 8, 2): 4 waves/block, 1 tile/wave
    dim3 bGemm(32, 4, 1);
    bernvae_in_gemm <<<gGemm, bGemm, 0, stream>>>(x, W1, W2, b1, b2, x1, x2);
    bernvae_lat_gemm<<<gGemm, bGemm, 0, stream>>>(x1, x2, Wp, bp, s1, s2p);

    dim3 gPair(NN / 16, NN / 16, 1);    // (32, 32)
    dim3 bPair(16, 16, 1);
    bernvae_pairwise<<<gPair, bPair, 0, stream>>>(s1, s2p, Wb, bb, noise, out);
}